// StockGNN_80229989089422
// MI455X (gfx1250) — compile-verified
//
#include <hip/hip_runtime.h>

typedef __attribute__((ext_vector_type(2))) float v2f;
typedef __attribute__((ext_vector_type(8))) float v8f;

#define D_HID 128

// ---------------------------------------------------------------------------
// utility: fast hardware f32 atomic add (global_atomic_add_f32)
// ---------------------------------------------------------------------------
__device__ __forceinline__ void fadd_atomic(float* p, float v) {
    unsafeAtomicAdd(p, v);
}

// ---------------------------------------------------------------------------
// zero a buffer (float4 granularity)
// ---------------------------------------------------------------------------
__global__ void zero4_kernel(float4* __restrict__ p, int n4) {
    int i = blockIdx.x * blockDim.x + threadIdx.x;
    if (i < n4) p[i] = make_float4(0.f, 0.f, 0.f, 0.f);
}

// ---------------------------------------------------------------------------
// degree: count incoming edges per node (self-loop added in dinv kernel)
// ---------------------------------------------------------------------------
__global__ void deg_kernel(const int* __restrict__ col, unsigned* __restrict__ deg, int nE) {
    int e = blockIdx.x * blockDim.x + threadIdx.x;
    if (e < nE) atomicAdd(&deg[col[e]], 1u);
}

__global__ void dinv_kernel(const unsigned* __restrict__ deg, float* __restrict__ dinv, int n) {
    int v = blockIdx.x * blockDim.x + threadIdx.x;
    if (v < n) dinv[v] = rsqrtf((float)(deg[v] + 1u));   // +1 self loop
}

// ---------------------------------------------------------------------------
// WMMA f32 GEMM:  C[M x NCOLS] = A[M x 128] * B[128 x NCOLS]  (+bias, +relu)
// NCOLS is compile-time so every B load is a single immediate-offset
// global_load_b32 from one per-lane base register (no u64 address math).
// blockDim.x = 32 * (NCOLS/16); wave w computes tile (blockIdx.x, w).
// A tile (16 x 128) staged in LDS, padded stride 132 (bank-conflict free).
// Requires M % 16 == 0 (100000 = 6250*16).
// ---------------------------------------------------------------------------
template <int NCOLS>
__global__ void wmma_gemm_f32(const float* __restrict__ A, const float* __restrict__ Bm,
                              float* __restrict__ C, const float* __restrict__ bias,
                              int relu) {
    __shared__ float As[16 * 132];
    const int m0  = blockIdx.x * 16;
    const int tid = threadIdx.x;

    // cooperative coalesced load of contiguous 16x128 A block (lda == 128)
    const float4* Asrc = (const float4*)(A + (size_t)m0 * D_HID);
    for (int i = tid; i < 16 * 32; i += blockDim.x) {
        int r = i >> 5, cq = i & 31;
        *(float4*)&As[r * 132 + cq * 4] = Asrc[i];
    }
    __syncthreads();

    const int lane   = tid & 31;
    const int wave   = tid >> 5;
    const int n0     = wave * 16;
    const int mlocal = lane & 15;          // A: row M, B: col N
    const int khalf  = (lane >> 4) * 2;    // lanes 16-31 hold K+2,K+3

    // per-lane bases; all inner-loop accesses are immediate offsets off these
    const float* Ap = &As[mlocal * 132 + khalf];
    const float* Bp = Bm + (size_t)khalf * NCOLS + n0 + mlocal;

    v8f acc = {};
#pragma unroll
    for (int k0 = 0; k0 < D_HID; k0 += 4) {
        v2f a, b;
        a[0] = Ap[k0];
        a[1] = Ap[k0 + 1];
        b[0] = Bp[k0 * NCOLS];
        b[1] = Bp[k0 * NCOLS + NCOLS];
        acc = __builtin_amdgcn_wmma_f32_16x16x4_f32(
            /*neg_a=*/false, a, /*neg_b=*/false, b,
            /*c_mod=*/(short)0, acc, /*reuse_a=*/false, /*reuse_b=*/false);
    }

    // D layout: VGPR r -> M = r (lanes 0-15) / r+8 (lanes 16-31); N = lane&15
    const int mbase = m0 + ((lane >> 4) << 3);
    const int nn    = n0 + mlocal;
    const float bv  = bias ? bias[nn] : 0.f;
#pragma unroll
    for (int r = 0; r < 8; ++r) {
        float v = acc[r] + bv;
        if (relu) v = fmaxf(v, 0.f);
        C[(size_t)(mbase + r) * NCOLS + nn] = v;
    }
}

// ---------------------------------------------------------------------------
// edge scatter: one wave per edge; lane handles 4 channels (float4 gather,
// 4 hardware f32 atomic adds into the L2-resident accumulator).
// ---------------------------------------------------------------------------
__global__ void edge_scatter(const float* __restrict__ t, const int* __restrict__ row,
                             const int* __restrict__ col, const float* __restrict__ dinv,
                             float* __restrict__ agg, int nE) {
    int e = blockIdx.x * (blockDim.x >> 5) + (threadIdx.x >> 5);
    if (e >= nE) return;
    int lane = threadIdx.x & 31;
    int r = row[e], c = col[e];
    float w = dinv[r] * dinv[c];
    float4 v = ((const float4*)(t + (size_t)r * D_HID))[lane];
    float* dst = agg + (size_t)c * D_HID + lane * 4;
    fadd_atomic(dst + 0, v.x * w);
    fadd_atomic(dst + 1, v.y * w);
    fadd_atomic(dst + 2, v.z * w);
    fadd_atomic(dst + 3, v.w * w);
}

// ---------------------------------------------------------------------------
// finalize: h = relu(BN(agg + t*dinv^2 + b))   (agg may alias h: elementwise)
// ---------------------------------------------------------------------------
__global__ void finalize_gcn(const float* __restrict__ t, const float* __restrict__ agg,
                             const float* __restrict__ dinv, const float* __restrict__ b,
                             const float* __restrict__ gamma, const float* __restrict__ beta,
                             const float* __restrict__ mean, const float* __restrict__ var,
                             float* __restrict__ h, int total) {
    int idx = blockIdx.x * blockDim.x + threadIdx.x;
    if (idx >= total) return;
    int v = idx >> 7, c = idx & 127;
    float d = dinv[v];
    float s = agg[idx] + t[idx] * d * d + b[c];
    s = (s - mean[c]) * rsqrtf(var[c] + 1e-5f) * gamma[c] + beta[c];
    h[idx] = fmaxf(s, 0.f);
}

// ---------------------------------------------------------------------------
// fc2: out[v] = dot(h[v, 0:64], w) + b ; one wave per node, shuffle reduce
// ---------------------------------------------------------------------------
__global__ void fc2_kernel(const float* __restrict__ h, const float* __restrict__ w,
                           const float* __restrict__ bp, float* __restrict__ out, int n) {
    int node = blockIdx.x * (blockDim.x >> 5) + (threadIdx.x >> 5);
    if (node >= n) return;
    int lane = threadIdx.x & 31;
    const float* hr = h + (size_t)node * 64;
    float s = hr[lane] * w[lane] + hr[lane + 32] * w[lane + 32];
    for (int off = 16; off; off >>= 1) s += __shfl_down(s, off, 32);
    if (lane == 0) out[node] = s + bp[0];
}

// ---------------------------------------------------------------------------
extern "C" void kernel_launch(void* const* d_in, const int* in_sizes, int n_in,
                              void* d_out, int out_size, void* d_ws, size_t ws_size,
                              hipStream_t stream) {
    const float* x    = (const float*)d_in[0];
    const int*   ei   = (const int*)d_in[1];     // [2, E]
    const float* W    = (const float*)d_in[2];   // [2, 128, 128]
    const float* bvec = (const float*)d_in[3];   // [2, 128]
    const float* gmm  = (const float*)d_in[4];
    const float* bet  = (const float*)d_in[5];
    const float* mu   = (const float*)d_in[6];
    const float* var  = (const float*)d_in[7];
    const float* fc1w = (const float*)d_in[8];   // [128, 64]
    const float* fc1b = (const float*)d_in[9];   // [64]
    const float* fc2w = (const float*)d_in[10];  // [64, 1]
    const float* fc2b = (const float*)d_in[11];  // [1]
    float* out = (float*)d_out;

    const int N = in_sizes[0] / D_HID;           // 100000
    const int E = in_sizes[1] / 2;               // 1600000

    // workspace: two [N,128] f32 ping-pong buffers + deg + dinv (~103 MB)
    size_t nb = (size_t)N * D_HID * sizeof(float);
    float*    B0   = (float*)d_ws;
    float*    B1   = (float*)((char*)d_ws + nb);
    unsigned* deg  = (unsigned*)((char*)d_ws + 2 * nb);
    float*    dinv = (float*)((char*)d_ws + 2 * nb + (size_t)N * sizeof(float));

    const int* row = ei;          // source
    const int* col = ei + E;      // target

    // --- normalization coefficients -------------------------------------
    int ndeg4 = N / 4;  // N multiple of 4
    zero4_kernel<<<(ndeg4 + 255) / 256, 256, 0, stream>>>((float4*)deg, ndeg4);
    deg_kernel<<<(E + 255) / 256, 256, 0, stream>>>(col, deg, E);
    dinv_kernel<<<(N + 255) / 256, 256, 0, stream>>>(deg, dinv, N);

    const int mtiles = N / 16;          // 6250
    const int feat4  = N * D_HID / 4;   // 3.2M float4

    // --- two GCN layers --------------------------------------------------
    for (int i = 0; i < 2; ++i) {
        const float* hin = (i == 0) ? x : B1;
        wmma_gemm_f32<128><<<mtiles, 256, 0, stream>>>(
            hin, W + (size_t)i * D_HID * D_HID, B0, nullptr, 0);
        zero4_kernel<<<(feat4 + 255) / 256, 256, 0, stream>>>((float4*)B1, feat4);
        edge_scatter<<<(E + 7) / 8, 256, 0, stream>>>(B0, row, col, dinv, B1, E);
        finalize_gcn<<<(N * D_HID + 255) / 256, 256, 0, stream>>>(
            B0, B1, dinv, bvec + i * D_HID, gmm + i * D_HID, bet + i * D_HID,
            mu + i * D_HID, var + i * D_HID, B1, N * D_HID);
    }

    // --- MLP head ---------------------------------------------------------
    // fc1: [N,128]x[128,64] with fused bias+relu (4 waves/block, NCOLS=64)
    wmma_gemm_f32<64><<<mtiles, 128, 0, stream>>>(B1, fc1w, B0, fc1b, 1);
    // fc2: [N,64]x[64,1]
    fc2_kernel<<<(N + 7) / 8, 256, 0, stream>>>(B0, fc2w, fc2b, out, N);
}